// CTE_37512244364031
// MI455X (gfx1250) — compile-verified
//
#include <hip/hip_runtime.h>
#include <cstdint>

typedef float v2f __attribute__((ext_vector_type(2)));
typedef float v8f __attribute__((ext_vector_type(8)));
typedef unsigned int v4u __attribute__((ext_vector_type(4)));
typedef unsigned int v8u __attribute__((ext_vector_type(8)));

#define M_F   8
#define K_F   12
#define L_F   9
#define DOUT  32
#define NB    64
#define CH    8
#define HH    128
#define WW    128
#define HPV   120
#define WPV   120
#define PO    7
#define OH    114
#define OW    114
#define OT    16              // output tile edge
#define VT    (OT + PO - 1)   // 22  votes tile edge
#define XT    (VT + L_F - 1)  // 30  x tile edge (TDM tile, contiguous rows)

// LDS x-tile: CH * XT * XT floats = 8*30*30*4 = 28800 B
// LDS votes half-slab (d-pair-major): 8 * 484 * 2 floats = 30976 B (overlays x-tile)
#define SMEM_U_BYTES (8 * VT * VT * 2 * 4)

__global__ __launch_bounds__(256)
void fern_fused(const float* __restrict__ x,
                const float* __restrict__ thrs,
                const float* __restrict__ table,
                const int*   __restrict__ chan,
                const int*   __restrict__ offs,
                float*       __restrict__ out)
{
    __shared__ __align__(16) unsigned char smem_u[SMEM_U_BYTES]; // xt / votes union
    __shared__ int            p_o1[M_F * K_F];
    __shared__ int            p_o2[M_F * K_F];
    __shared__ float          p_thr[M_F * K_F];
    __shared__ unsigned short words_s[M_F * VT * VT];
    __shared__ float          conf_s[M_F * VT * VT];

    const int n   = blockIdx.z;
    const int oy0 = blockIdx.y * OT;
    const int ox0 = blockIdx.x * OT;
    const int tid = threadIdx.x;
    const int wave = tid >> 5;
    const int lane = tid & 31;

    // ---- Phase A0: fern parameters -> LDS ----------------------------------
    for (int i = tid; i < M_F * K_F; i += 256) {
        int c1  = chan[i * 2 + 0];
        int c2  = chan[i * 2 + 1];
        int dy1 = offs[i * 4 + 0];
        int dx1 = offs[i * 4 + 1];
        int dy2 = offs[i * 4 + 2];
        int dx2 = offs[i * 4 + 3];
        p_o1[i]  = (c1 * XT + dy1) * XT + dx1;
        p_o2[i]  = (c2 * XT + dy2) * XT + dx2;
        p_thr[i] = thrs[i];
    }

    // ---- Phase A1: TDM tensor load of x tile (30x30x8 f32) into LDS --------
    // 3D tile, tensor dims given as remaining extent from tile start so the
    // TDM zero-fills out-of-bounds rows/cols (keeps NaNs out of the WMMA).
    if (wave == 0) {
        const unsigned ldsXt = (unsigned)(uintptr_t)(&smem_u[0]);
        const uint64_t ga = (uint64_t)(uintptr_t)x +
                            ((uint64_t)n * (CH * HH * WW) +
                             (uint64_t)oy0 * WW + (uint64_t)ox0) * 4u;
        const unsigned td0 = (unsigned)(WW - ox0);   // remaining cols (elements)
        const unsigned td1 = (unsigned)(HH - oy0);   // remaining rows

        // D# group 0: count=1 | lds_addr | global_addr | type=2
        v4u g0 = { 1u,
                   ldsXt,
                   (unsigned)(ga & 0xFFFFFFFFu),
                   (unsigned)((ga >> 32) & 0x1FFFFFFu) | (2u << 30) };
        // D# group 1: wg_mask=0, data_size=4B, dims/strides/tile dims
        v8u g1 = { (2u << 16),                          // data_size = 2 (4 bytes)
                   (td0 & 0xFFFFu) << 16,               // tensor_dim0 lo16 @ [63:48]
                   (td0 >> 16) | ((td1 & 0xFFFFu) << 16), // tensor_dim0 hi / dim1 lo
                   (td1 >> 16) | ((unsigned)XT << 16),  // tensor_dim1 hi / tile_dim0=30
                   (unsigned)XT | ((unsigned)CH << 16), // tile_dim1=30, tile_dim2=8
                   (unsigned)WW,                        // tensor_dim0_stride = 128
                   ((unsigned)(HH * WW) & 0xFFFFu) << 16, // dim0_str hi=0 | dim1_str lo16
                   (unsigned)((HH * WW) >> 16) };       // dim1_stride hi
        // D# group 2: tensor_dim2=8 | dim3=0 | tensor_dim2_stride=128*128 | tile_dim3=0
        v4u g2 = { (unsigned)CH, 0u, (unsigned)(HH * WW), 0u };
        // D# group 3: unused dims
        v4u g3 = { 0u, 0u, 0u, 0u };

        asm volatile("tensor_load_to_lds %0, %1, %2, %3"
                     :
                     : "s"(g0), "s"(g1), "s"(g2), "s"(g3)
                     : "memory");
        __builtin_amdgcn_s_wait_tensorcnt(0);
    }
    __syncthreads();

    // ---- Phase B: words + confidences per pixel ----------------------------
    {
        const float* xt = (const float*)smem_u;
        for (int px = tid; px < VT * VT; px += 256) {
            int vy = px / VT;
            int vx = px - vy * VT;
            bool valid = ((oy0 + vy) < HPV) && ((ox0 + vx) < WPV);
            int pb = vy * XT + vx;
            for (int m = 0; m < M_F; ++m) {
                unsigned w = 0;
                float cf = 1.0f;
                if (valid) {
#pragma unroll
                    for (int k = 0; k < K_F; ++k) {
                        int i = m * K_F + k;
                        float dff = xt[p_o1[i] + pb] - xt[p_o2[i] + pb];
                        float z = dff - p_thr[i];
                        unsigned bit = (z > 0.0f) ? 1u : 0u;
                        w |= bit << k;
                        // where(bit, s, 1-s) == sigmoid(|z|/TEMP), TEMP = 0.1
                        cf *= 1.0f / (1.0f + __expf(-fabsf(z) * 10.0f));
                    }
                } else {
                    cf = 0.0f;
                }
                words_s[m * (VT * VT) + px] = (unsigned short)w;
                conf_s [m * (VT * VT) + px] = cf;
            }
        }
    }
    __syncthreads();   // also guards xt -> votes LDS reuse

    // ---- Phase C/D: per d-half, votes slab then WMMA-pooled store ----------
    // votes layout: [d-pair p][px][j], j in {0,1}; p = local_d >> 1.
    float* votes = (float*)smem_u;           // overlays xt
    const int hi   = lane >> 4;              // half-wave select
    const int mrow = lane & 15;              // out row (A) / out col (B,C)

    for (int dh = 0; dh < 2; ++dh) {
        // votes = sum_m conf * table[m][word][dh*16 + dlocal]
        for (int px = tid; px < VT * VT; px += 256) {
            float acc[16];
#pragma unroll
            for (int j = 0; j < 16; ++j) acc[j] = 0.0f;
            for (int m = 0; m < M_F; ++m) {
                float    cf = conf_s [m * (VT * VT) + px];
                unsigned w  = words_s[m * (VT * VT) + px];
                const float* row = table + (((size_t)m << 12) + w) * DOUT + dh * 16;
#pragma unroll
                for (int q = 0; q < 4; ++q) {
                    float4 rv = ((const float4*)row)[q];
                    acc[4 * q + 0] += cf * rv.x;
                    acc[4 * q + 1] += cf * rv.y;
                    acc[4 * q + 2] += cf * rv.z;
                    acc[4 * q + 3] += cf * rv.w;
                }
            }
#pragma unroll
            for (int p = 0; p < 8; ++p) {
                float2 st; st.x = acc[2 * p]; st.y = acc[2 * p + 1];
                *(float2*)(votes + p * (VT * VT * 2) + px * 2) = st;
            }
        }
        __syncthreads();

        // Vertical 7-pool as banded 16x16x4 f32 WMMA; horizontal 7-sum on the fly.
        // Each wave owns d-pair `wave`: d = dh*16 + wave*2 + {0,1}.
        const float* vp = votes + wave * (VT * VT * 2);
        v8f c0 = {};
        v8f c1 = {};
#pragma unroll
        for (int kc = 0; kc < 6; ++kc) {
            v2f a, b0, b1;
#pragma unroll
            for (int v = 0; v < 2; ++v) {
                int kk = kc * 4 + v + hi * 2;                 // K index = vy (0..23)
                a[v] = (kk >= mrow && kk <= mrow + 6) ? 1.0f : 0.0f;
                int kkc = (kk < VT) ? kk : 0;
                float s0 = 0.0f, s1 = 0.0f;
#pragma unroll
                for (int t = 0; t < PO; ++t) {
                    const float2 hv = *(const float2*)(vp + (kkc * VT + mrow + t) * 2);
                    s0 += hv.x;
                    s1 += hv.y;
                }
                float msk = (kk < VT) ? 1.0f : 0.0f;
                b0[v] = s0 * msk;
                b1[v] = s1 * msk;
            }
            c0 = __builtin_amdgcn_wmma_f32_16x16x4_f32(false, a, false, b0,
                                                       (short)0, c0, false, false);
            c1 = __builtin_amdgcn_wmma_f32_16x16x4_f32(false, a, false, b1,
                                                       (short)0, c1, false, false);
        }

        const int d0 = dh * 16 + wave * 2;
        const float inv = 1.0f / (float)(PO * PO);
#pragma unroll
        for (int r = 0; r < 8; ++r) {
            int oyl = r + hi * 8;                  // C layout: M = vgpr + 8*(lane>=16)
            int oy  = oy0 + oyl;
            int ox  = ox0 + mrow;                  // C layout: N = lane & 15
            if (oy < OH && ox < OW) {
                size_t base = ((size_t)(n * DOUT + d0)) * (OH * OW) + (size_t)oy * OW + ox;
                out[base]             = c0[r] * inv;
                out[base + (OH * OW)] = c1[r] * inv;
            }
        }
        __syncthreads();   // votes slab reused next dh
    }
}

extern "C" void kernel_launch(void* const* d_in, const int* in_sizes, int n_in,
                              void* d_out, int out_size, void* d_ws, size_t ws_size,
                              hipStream_t stream) {
    const float* x     = (const float*)d_in[0];
    const float* thrs  = (const float*)d_in[1];
    const float* table = (const float*)d_in[2];
    const int*   chan  = (const int*)d_in[3];
    const int*   offs  = (const int*)d_in[4];
    float*       out   = (float*)d_out;

    dim3 grid((OW + OT - 1) / OT, (OH + OT - 1) / OT, NB);   // 8 x 8 x 64
    fern_fused<<<grid, dim3(256), 0, stream>>>(x, thrs, table, chan, offs, out);
    (void)in_sizes; (void)n_in; (void)out_size; (void)d_ws; (void)ws_size;
}